// Quantizer_42992622633493
// MI455X (gfx1250) — compile-verified
//
#include <hip/hip_runtime.h>

typedef __attribute__((ext_vector_type(2))) float v2f;
typedef __attribute__((ext_vector_type(8))) float v8f;

#define N_POS   8192      // B*H*W = 8*32*32
#define IN_CH   1024
#define EMB     256
#define N_EMB   16384
#define HW      1024      // 32*32
#define LDSK    (EMB + 4) // padded LDS row stride (floats) -> rotates banks per row
#define MB      32        // z rows per block in the argmin kernel

// ---------------------------------------------------------------------------
// Kernel 1: z[m,e] = sum_c feats[b,c,hw] * conv_w[e,c] + conv_b[e]
// One wave per 16x16 output tile, K looped in steps of 4 via v_wmma_f32_16x16x4_f32.
// ---------------------------------------------------------------------------
__global__ void k_conv_gemm(const float* __restrict__ feats,
                            const float* __restrict__ conv_w,
                            const float* __restrict__ conv_b,
                            float* __restrict__ z)
{
    const int wave = threadIdx.x >> 5;
    const int lane = threadIdx.x & 31;
    const int tile = blockIdx.x * 8 + wave;   // 8192 tiles total
    const int mt = tile >> 4;                 // 0..511  (rows of 16)
    const int et = tile & 15;                 // 0..15   (cols of 16)
    const int m0 = mt * 16, e0 = et * 16;
    const int half = lane >> 4;               // 0/1
    const int mr   = lane & 15;               // 0..15

    // A-fragment source: row m = m0+mr of z-input; feats layout [B,C,HW]
    const int m  = m0 + mr;
    const int b  = m >> 10;
    const int hw = m & (HW - 1);
    const float* fbase = feats + (size_t)b * IN_CH * HW + hw;   // +c*HW
    // B-fragment source: column e = e0+mr; conv_w layout [EMB, IN_CH]
    const float* wbase = conv_w + (size_t)(e0 + mr) * IN_CH;    // +k

    v8f acc = {};
    for (int kk = 0; kk < IN_CH; kk += 4) {
        const int ka = kk + 2 * half;
        v2f a, bf;
        a.x = fbase[(size_t)ka * HW];            // A[mr][ka]
        a.y = fbase[(size_t)(ka + 1) * HW];      // A[mr][ka+1]
        bf  = *(const v2f*)(wbase + ka);         // B[ka..ka+1][e0+mr] (contig)
        acc = __builtin_amdgcn_wmma_f32_16x16x4_f32(
                  false, a, false, bf, (short)0, acc, false, false);
    }
    const float bias = conv_b[e0 + mr];
    #pragma unroll
    for (int v = 0; v < 8; ++v) {
        const int row = m0 + v + 8 * half;       // C/D layout: M = v + 8*half, N = mr
        z[(size_t)row * EMB + (e0 + mr)] = acc[v] + bias;
    }
}

// ---------------------------------------------------------------------------
// Kernel 2a: ce[n] = 0.5 * ||codebook[n]||^2
// ---------------------------------------------------------------------------
__global__ void k_codebook_norm(const float* __restrict__ cb,
                                float* __restrict__ ce)
{
    const int n = blockIdx.x * blockDim.x + threadIdx.x;
    if (n >= N_EMB) return;
    const float* row = cb + (size_t)n * EMB;
    float s = 0.f;
    #pragma unroll 8
    for (int k = 0; k < EMB; ++k) s += row[k] * row[k];
    ce[n] = 0.5f * s;
}

// ---------------------------------------------------------------------------
// Kernel 2b: fused score-GEMM + argmax, 2x2 register blocked.
// Block = 32 rows of z staged in LDS (halves L2 codebook traffic vs 16 rows).
// Each wave computes a 32x32 score tile per iteration: 4 accumulators reuse
// 2 A-fragments (LDS) and 2 B-fragments (codebook rows, L2-resident), so the
// inner loop issues 4 v_wmma per 4 loads. score = z.e - 0.5||e||^2; argmax of
// score == argmin of the full squared distance (row term is constant).
// ---------------------------------------------------------------------------
__global__ void k_argmin(const float* __restrict__ z,
                         const float* __restrict__ cb,
                         const float* __restrict__ ce,
                         int* __restrict__ idx)
{
    __shared__ float lz[MB * LDSK];
    __shared__ float s_best[MB][8];
    __shared__ int   s_idx[MB][8];

    const int m0 = blockIdx.x * MB;
    // stage 32 x 256 z tile into LDS (padded rows)
    for (int t = threadIdx.x; t < MB * EMB; t += blockDim.x) {
        const int r = t >> 8;
        const int c = t & 255;
        lz[r * LDSK + c] = z[(size_t)(m0 + r) * EMB + c];
    }
    __syncthreads();

    const int wave = threadIdx.x >> 5;
    const int lane = threadIdx.x & 31;
    const int half = lane >> 4;
    const int mr   = lane & 15;

    float bestS[2][8];
    int   bestN[2][8];
    #pragma unroll
    for (int g = 0; g < 2; ++g)
        #pragma unroll
        for (int v = 0; v < 8; ++v) { bestS[g][v] = -3.4e38f; bestN[g][v] = 0; }

    const float* lzA0 = &lz[mr * LDSK];
    const float* lzA1 = &lz[(16 + mr) * LDSK];

    for (int nt = wave; nt < N_EMB / 32; nt += 8) {  // 512 col-tiles of 32, 64/wave
        const int n0 = nt * 32;
        const float* brow0 = cb + (size_t)(n0 + mr) * EMB;
        const float* brow1 = cb + (size_t)(n0 + 16 + mr) * EMB;
        // prefetch this wave's next tile (+8*32 codebook rows = 256 KB ahead)
        __builtin_prefetch(brow0 + (size_t)256 * EMB, 0, 0);
        __builtin_prefetch(brow1 + (size_t)256 * EMB, 0, 0);

        v8f acc00 = {}, acc01 = {}, acc10 = {}, acc11 = {};
        #pragma unroll 4
        for (int kk = 0; kk < EMB; kk += 4) {
            const int ka = kk + 2 * half;
            const v2f a0 = *(const v2f*)(lzA0 + ka);   // rows m0..m0+15
            const v2f a1 = *(const v2f*)(lzA1 + ka);   // rows m0+16..m0+31
            const v2f b0 = *(const v2f*)(brow0 + ka);  // cols n0..n0+15
            const v2f b1 = *(const v2f*)(brow1 + ka);  // cols n0+16..n0+31
            acc00 = __builtin_amdgcn_wmma_f32_16x16x4_f32(
                        false, a0, false, b0, (short)0, acc00, false, false);
            acc01 = __builtin_amdgcn_wmma_f32_16x16x4_f32(
                        false, a0, false, b1, (short)0, acc01, false, false);
            acc10 = __builtin_amdgcn_wmma_f32_16x16x4_f32(
                        false, a1, false, b0, (short)0, acc10, false, false);
            acc11 = __builtin_amdgcn_wmma_f32_16x16x4_f32(
                        false, a1, false, b1, (short)0, acc11, false, false);
        }
        const float ce0 = ce[n0 + mr];
        const float ce1 = ce[n0 + 16 + mr];
        const int   nc0 = n0 + mr;
        const int   nc1 = n0 + 16 + mr;
        #pragma unroll
        for (int v = 0; v < 8; ++v) {
            // check smaller-n candidate first; strict > keeps first index on ties
            float s = acc00[v] - ce0;
            if (s > bestS[0][v]) { bestS[0][v] = s; bestN[0][v] = nc0; }
            s = acc01[v] - ce1;
            if (s > bestS[0][v]) { bestS[0][v] = s; bestN[0][v] = nc1; }
            s = acc10[v] - ce0;
            if (s > bestS[1][v]) { bestS[1][v] = s; bestN[1][v] = nc0; }
            s = acc11[v] - ce1;
            if (s > bestS[1][v]) { bestS[1][v] = s; bestN[1][v] = nc1; }
        }
    }

    // reduce across the 16 lanes holding each row (xor 1,2,4,8 stays in 16-group)
    #pragma unroll
    for (int g = 0; g < 2; ++g) {
        #pragma unroll
        for (int v = 0; v < 8; ++v) {
            float s = bestS[g][v];
            int   n = bestN[g][v];
            #pragma unroll
            for (int off = 1; off < 16; off <<= 1) {
                const float so = __shfl_xor(s, off, 32);
                const int   no = __shfl_xor(n, off, 32);
                if (so > s || (so == s && no < n)) { s = so; n = no; }
            }
            if (mr == 0) {
                const int row = g * 16 + v + 8 * half;   // C/D: M = v + 8*half
                s_best[row][wave] = s;
                s_idx [row][wave] = n;
            }
        }
    }
    __syncthreads();

    if (threadIdx.x < MB) {
        float s = s_best[threadIdx.x][0];
        int   n = s_idx [threadIdx.x][0];
        #pragma unroll
        for (int w = 1; w < 8; ++w) {
            const float so = s_best[threadIdx.x][w];
            const int   no = s_idx [threadIdx.x][w];
            if (so > s || (so == s && no < n)) { s = so; n = no; }
        }
        idx[m0 + threadIdx.x] = n;
    }
}

// ---------------------------------------------------------------------------
// Kernel 3: gather z_q = codebook[idx], write transposed [B,E,H,W] output,
// and accumulate deterministic per-block partial loss sums.
// ---------------------------------------------------------------------------
__global__ void k_gather(const float* __restrict__ z,
                         const float* __restrict__ cb,
                         const int* __restrict__ idx,
                         float* __restrict__ out,
                         float* __restrict__ partial)
{
    __shared__ float red[256];
    const int o  = blockIdx.x * 256 + threadIdx.x;   // output element, coalesced
    const int b  = o >> 18;          // / (EMB*HW)
    const int e  = (o >> 10) & 255;
    const int hw = o & 1023;
    const int m  = (b << 10) | hw;
    const int n  = idx[m];
    const float zq = cb[(size_t)n * EMB + e];
    const float zv = z[(size_t)m * EMB + e];
    out[o] = zq;                                     // straight-through fwd == z_q
    const float d = zq - zv;
    red[threadIdx.x] = 1.25f * d * d;                // (1 + BETA) * (z_q - z)^2
    __syncthreads();
    for (int s = 128; s > 0; s >>= 1) {
        if (threadIdx.x < s) red[threadIdx.x] += red[threadIdx.x + s];
        __syncthreads();
    }
    if (threadIdx.x == 0) partial[blockIdx.x] = red[0];
}

// ---------------------------------------------------------------------------
// Kernel 4: deterministic final loss reduction (8192 partials -> 1 scalar)
// ---------------------------------------------------------------------------
__global__ void k_loss(const float* __restrict__ partial,
                       float* __restrict__ loss_out)
{
    __shared__ float red[256];
    float s = 0.f;
    for (int i = threadIdx.x; i < 8192; i += 256) s += partial[i];
    red[threadIdx.x] = s;
    __syncthreads();
    for (int st = 128; st > 0; st >>= 1) {
        if (threadIdx.x < st) red[threadIdx.x] += red[threadIdx.x + st];
        __syncthreads();
    }
    if (threadIdx.x == 0) loss_out[0] = red[0] * (1.0f / 2097152.0f);
}

// ---------------------------------------------------------------------------
extern "C" void kernel_launch(void* const* d_in, const int* in_sizes, int n_in,
                              void* d_out, int out_size, void* d_ws, size_t ws_size,
                              hipStream_t stream)
{
    const float* feats    = (const float*)d_in[0];  // [8,1024,32,32]
    const float* conv_w   = (const float*)d_in[1];  // [256,1024]
    const float* conv_b   = (const float*)d_in[2];  // [256]
    const float* codebook = (const float*)d_in[3];  // [16384,256]
    float* out = (float*)d_out;                     // [2097152 z_q] + [1 loss]

    char* ws = (char*)d_ws;
    float* z       = (float*)(ws);                                    // 8 MB
    float* ce      = (float*)(ws + (size_t)8 * 1024 * 1024);          // 64 KB
    int*   idx     = (int*)  (ws + (size_t)8 * 1024 * 1024 + 65536);  // 32 KB
    float* partial = (float*)(ws + (size_t)8 * 1024 * 1024 + 65536 + 32768); // 32 KB

    k_conv_gemm    <<<1024, 256, 0, stream>>>(feats, conv_w, conv_b, z);
    k_codebook_norm<<<  64, 256, 0, stream>>>(codebook, ce);
    k_argmin       <<< 256, 256, 0, stream>>>(z, codebook, ce, idx);
    k_gather       <<<8192, 256, 0, stream>>>(z, codebook, idx, out, partial);
    k_loss         <<<   1, 256, 0, stream>>>(partial, out + (out_size - 1));
}